// MultiHeadAttentionLayer_47493748359302
// MI455X (gfx1250) — compile-verified
//
#include <hip/hip_runtime.h>

#define NNODES 50000
#define IN_DIM 128
#define HD     64      // H*D
#define NH     8
#define DH     8
#define NEDGES 800000

typedef __attribute__((ext_vector_type(2))) float v2f;
typedef __attribute__((ext_vector_type(8))) float v8f;

// ---------------------------------------------------------------- zero fill
__global__ void mha_zero(float* __restrict__ p, int n) {
    int i = blockIdx.x * blockDim.x + threadIdx.x;
    if (i < n) p[i] = 0.0f;
}

// ---------------------------------------------------------------- QKV GEMM
// One wave32 computes one 16x16 tile of Out = h(50000x128) @ W(128x64) + b.
// 4 waves per block cover the 64 output columns; blockIdx.y picks Q/K/V.
// 50000 % 16 == 0, so no row guards -> EXEC all-1s for WMMA.
__global__ __launch_bounds__(128) void mha_qkv_gemm(
    const float* __restrict__ h,
    const float* __restrict__ Wq, const float* __restrict__ bq,
    const float* __restrict__ Wk, const float* __restrict__ bk,
    const float* __restrict__ Wv, const float* __restrict__ bv,
    float* __restrict__ Qo, float* __restrict__ Ko, float* __restrict__ Vo)
{
    const float* W; const float* bias; float* Out;
    if (blockIdx.y == 0)      { W = Wq; bias = bq; Out = Qo; }
    else if (blockIdx.y == 1) { W = Wk; bias = bk; Out = Ko; }
    else                      { W = Wv; bias = bv; Out = Vo; }

    const int wave = threadIdx.x >> 5;    // 0..3 -> 16-col tile
    const int lane = threadIdx.x & 31;
    const int half = lane >> 4;           // lane group selects K pair
    const int lm   = lane & 15;
    const int row0 = blockIdx.x * 16;
    const int col0 = wave * 16;

    // A frag: lane holds A[M=lm][kk+2*half .. +1]  (contiguous -> b64 load)
    const float* arow = h + (size_t)(row0 + lm) * IN_DIM + 2 * half;
    // B frag: lane holds B[kk+2*half .. +1][N=col0+lm]  (stride HD)
    const float* bcol = W + (size_t)(2 * half) * HD + col0 + lm;

    v8f acc = {};
    for (int kk = 0; kk < IN_DIM; kk += 4) {
        v2f a = *reinterpret_cast<const v2f*>(arow + kk);   // 8B aligned
        const float* bp = bcol + (size_t)kk * HD;
        v2f b = { bp[0], bp[HD] };
        // D = A(16x4,f32) * B(4x16,f32) + C   -> v_wmma_f32_16x16x4_f32
        acc = __builtin_amdgcn_wmma_f32_16x16x4_f32(
            false, a, false, b, (short)0, acc, false, false);
    }

    const float bb = bias[col0 + lm];
    // D layout: VGPR v holds M = v + 8*half, N = lm
    #pragma unroll
    for (int v = 0; v < 8; ++v) {
        int m = row0 + v + 8 * half;
        Out[(size_t)m * HD + col0 + lm] = acc[v] + bb;
    }
}

// ---------------------------------------------------------------- edge phase
// One thread per (edge, head). 8 consecutive threads share src/dst, so the
// 64-float K/V/Q row reads coalesce into full cachelines (L2-resident: QKV
// totals 38.4 MB << 192 MB L2). Scatter via global_atomic_add_f32.
__global__ __launch_bounds__(256) void mha_edge(
    const float* __restrict__ Q, const float* __restrict__ K,
    const float* __restrict__ V,
    const int* __restrict__ src, const int* __restrict__ dst,
    float* __restrict__ wV, float* __restrict__ z)
{
    int t = blockIdx.x * blockDim.x + threadIdx.x;
    if (t >= NEDGES * NH) return;
    const int e  = t >> 3;
    const int hh = t & 7;
    const int s  = src[e];
    const int d  = dst[e];
    const size_t so = (size_t)s * HD + hh * DH;
    const size_t dо = (size_t)d * HD + hh * DH;

    float4 k0 = *reinterpret_cast<const float4*>(K + so);
    float4 k1 = *reinterpret_cast<const float4*>(K + so + 4);
    float4 q0 = *reinterpret_cast<const float4*>(Q + dо);
    float4 q1 = *reinterpret_cast<const float4*>(Q + dо + 4);

    float sc = k0.x*q0.x + k0.y*q0.y + k0.z*q0.z + k0.w*q0.w
             + k1.x*q1.x + k1.y*q1.y + k1.z*q1.z + k1.w*q1.w;
    sc *= 0.35355339059327373f;                 // 1 / sqrt(8)
    sc  = fminf(5.0f, fmaxf(-5.0f, sc));
    sc  = __expf(sc);

    float4 v0 = *reinterpret_cast<const float4*>(V + so);
    float4 v1 = *reinterpret_cast<const float4*>(V + so + 4);

    float* o = wV + dо;
    atomicAdd(o + 0, v0.x * sc); atomicAdd(o + 1, v0.y * sc);
    atomicAdd(o + 2, v0.z * sc); atomicAdd(o + 3, v0.w * sc);
    atomicAdd(o + 4, v1.x * sc); atomicAdd(o + 5, v1.y * sc);
    atomicAdd(o + 6, v1.z * sc); atomicAdd(o + 7, v1.w * sc);
    atomicAdd(z + (size_t)d * NH + hh, sc);
}

// ---------------------------------------------------------------- normalize
__global__ void mha_finalize(const float* __restrict__ wV,
                             const float* __restrict__ z,
                             float* __restrict__ out)
{
    int t = blockIdx.x * blockDim.x + threadIdx.x;
    if (t >= NNODES * HD) return;
    const int n  = t >> 6;
    const int hh = (t >> 3) & 7;
    out[t] = wV[t] / z[(size_t)n * NH + hh];
}

// ---------------------------------------------------------------- launcher
extern "C" void kernel_launch(void* const* d_in, const int* in_sizes, int n_in,
                              void* d_out, int out_size, void* d_ws, size_t ws_size,
                              hipStream_t stream)
{
    const float* h  = (const float*)d_in[0];
    const float* Wq = (const float*)d_in[1];
    const float* bq = (const float*)d_in[2];
    const float* Wk = (const float*)d_in[3];
    const float* bk = (const float*)d_in[4];
    const float* Wv = (const float*)d_in[5];
    const float* bv = (const float*)d_in[6];
    const int*  src = (const int*)d_in[7];
    const int*  dst = (const int*)d_in[8];
    float* out = (float*)d_out;

    // Workspace layout (fp32): Q | K | V | wV | z   = 50000*(4*64+8)*4 B ~ 52.8 MB
    float* Q  = (float*)d_ws;
    float* K  = Q  + (size_t)NNODES * HD;
    float* V  = K  + (size_t)NNODES * HD;
    float* wV = V  + (size_t)NNODES * HD;
    float* z  = wV + (size_t)NNODES * HD;

    // 1) zero accumulators (wV and z are contiguous)
    const int zn = NNODES * (HD + NH);
    mha_zero<<<(zn + 255) / 256, 256, 0, stream>>>(wV, zn);

    // 2) Q/K/V projections via v_wmma_f32_16x16x4_f32
    dim3 grid(NNODES / 16, 3);
    mha_qkv_gemm<<<grid, 128, 0, stream>>>(h, Wq, bq, Wk, bk, Wv, bv, Q, K, V);

    // 3) edge attention + scatter
    const int et = NEDGES * NH;
    mha_edge<<<(et + 255) / 256, 256, 0, stream>>>(Q, K, V, src, dst, wV, z);

    // 4) normalize
    const int ft = NNODES * HD;
    mha_finalize<<<(ft + 255) / 256, 256, 0, stream>>>(wV, z, out);
}